// GodSlayer2_14147622273220
// MI455X (gfx1250) — compile-verified
//
#include <hip/hip_runtime.h>
#include <math.h>

typedef __attribute__((ext_vector_type(2))) float v2f;
typedef __attribute__((ext_vector_type(8))) float v8f;

#define LL 128
#define NCH 8
#define NLAYERS 16

__device__ __forceinline__ float celu_f(float y) {
    // celu(y) = max(y,0) + expm1(min(y,0))
    return fmaxf(y, 0.0f) + expm1f(fminf(y, 0.0f));
}

__global__ __launch_bounds__(512)
void gram_stencil_kernel(const float* __restrict__ nsrc,
                         const float* __restrict__ wsrc,
                         float* __restrict__ partial)
{
    __shared__ float xbuf[LL * LL];   // 64 KB, resident state for this (b,c)

    const int bc   = blockIdx.x;      // 0..2047
    const int b    = bc >> 3;
    const int ch   = bc & 7;
    const int t    = threadIdx.x;     // 0..511
    const int lane = t & 31;
    const int wv   = t >> 5;          // 0..15 waves
    const int ls   = lane & 15;
    const bool hi  = (lane >= 16);

    // ---------------- Phase 1: g = n nT via V_WMMA_F32_16X16X4_F32 ----------
    // C tile (I,J): C[m][nn] = sum_{k<3} n[I*16+m][k] * n[J*16+nn][k]
    // A 16x4 f32 layout: lanes 0-15 hold {K0,K1}, lanes 16-31 hold {K2,K3};
    // B 4x16 layout mirrors it row-striped -> identical per-lane data (g symmetric).
    // Branchless fragment build: two unconditional in-bounds loads + value select.
    {
        const float* nb = nsrc + b * (LL * 3);
        const int I = wv >> 1;                  // each wave: one tile row, 4 cols
        {
        }
        const int ra   = I * 16 + ls;
        const int ra0  = ra * 3 + (hi ? 2 : 0); // x component address
        const int ra1  = ra * 3 + (hi ? 0 : 1); // y component address (dummy for hi)
        const float av0 = nb[ra0];
        const float av1 = nb[ra1];
        v2f aF;
        aF.x = av0;
        aF.y = hi ? 0.0f : av1;                 // K=3 column is zero
        #pragma unroll
        for (int jj = 0; jj < 4; ++jj) {
            const int J   = (wv & 1) * 4 + jj;
            const int rb  = J * 16 + ls;
            const int rb0 = rb * 3 + (hi ? 2 : 0);
            const int rb1 = rb * 3 + (hi ? 0 : 1);
            const float bv0 = nb[rb0];
            const float bv1 = nb[rb1];
            v2f bF;
            bF.x = bv0;
            bF.y = hi ? 0.0f : bv1;
            v8f acc = {};
            acc = __builtin_amdgcn_wmma_f32_16x16x4_f32(
                false, aF, false, bF, (short)0, acc, false, false);
            const int j = J * 16 + ls;
            #pragma unroll
            for (int v = 0; v < 8; ++v) {
                const int i = I * 16 + v + (hi ? 8 : 0);
                float val = acc[v];
                if (i == j) val = 0.0f;        // zero diagonal of g (cndmask)
                xbuf[i * LL + j] = val;
            }
        }
    }
    __syncthreads();

    // ---------------- Phase 2: 16 depthwise cross-stencil layers -------------
    // Each thread owns two 4x4 blocks (1024 blocks total, 32x32 grid).
    float zst[2][16];                           // staged x_new, written after barrier
    for (int l = 0; l < NLAYERS; ++l) {
        const float* wl = wsrc + (l * NCH + ch) * 5;
        float wd[5];
        #pragma unroll
        for (int d = 0; d < 5; ++d) wd[d] = wl[d];

        #pragma unroll
        for (int sb = 0; sb < 2; ++sb) {
            const int blk = t + sb * 512;
            const int i0 = (blk >> 5) * 4;
            const int j0 = (blk & 31) * 4;

            // horizontal window: 4 rows x 12 cols (3 aligned float4 per row, wrap via &127)
            float hw[4][12];
            #pragma unroll
            for (int r = 0; r < 4; ++r) {
                const float* rowp = &xbuf[(i0 + r) * LL];
                const float4 a = *(const float4*)(rowp + ((j0 - 4) & (LL - 1)));
                const float4 m = *(const float4*)(rowp + j0);
                const float4 p = *(const float4*)(rowp + ((j0 + 4) & (LL - 1)));
                hw[r][0] = a.x; hw[r][1]  = a.y; hw[r][2]  = a.z; hw[r][3]  = a.w;
                hw[r][4] = m.x; hw[r][5]  = m.y; hw[r][6]  = m.z; hw[r][7]  = m.w;
                hw[r][8] = p.x; hw[r][9]  = p.y; hw[r][10] = p.z; hw[r][11] = p.w;
            }
            // vertical halos: rows i0-4..i0-1 and i0+4..i0+7, cols j0..j0+3
            float vup[4][4], vdn[4][4];
            #pragma unroll
            for (int r = 0; r < 4; ++r) {
                const float4 u = *(const float4*)(&xbuf[((i0 - 4 + r) & (LL - 1)) * LL + j0]);
                const float4 d = *(const float4*)(&xbuf[((i0 + 4 + r) & (LL - 1)) * LL + j0]);
                vup[r][0] = u.x; vup[r][1] = u.y; vup[r][2] = u.z; vup[r][3] = u.w;
                vdn[r][0] = d.x; vdn[r][1] = d.y; vdn[r][2] = d.z; vdn[r][3] = d.w;
            }
            #pragma unroll
            for (int r = 0; r < 4; ++r) {
                #pragma unroll
                for (int cc = 0; cc < 4; ++cc) {
                    const float center = hw[r][4 + cc];
                    float acc = wd[0] * center;
                    #pragma unroll
                    for (int d = 1; d <= 4; ++d) {
                        const float hsum = hw[r][4 + cc - d] + hw[r][4 + cc + d];
                        const int ru = r - d;
                        const int rd = r + d;
                        const float vu = (ru >= 0) ? hw[ru][4 + cc] : vup[ru + 4][cc];
                        const float vv = (rd <= 3) ? hw[rd][4 + cc] : vdn[rd - 4][cc];
                        acc += wd[d] * (hsum + vu + vv);
                    }
                    // stage full new value: writeback is store-only (RESIDUAL_SCALE==1)
                    zst[sb][r * 4 + cc] = center + celu_f(acc);
                }
            }
        }
        __syncthreads();           // all reads of layer-l state done
        #pragma unroll
        for (int sb = 0; sb < 2; ++sb) {
            const int blk = t + sb * 512;
            const int i0 = (blk >> 5) * 4;
            const int j0 = (blk & 31) * 4;
            #pragma unroll
            for (int r = 0; r < 4; ++r) {
                float4 outv;
                outv.x = zst[sb][r * 4 + 0];
                outv.y = zst[sb][r * 4 + 1];
                outv.z = zst[sb][r * 4 + 2];
                outv.w = zst[sb][r * 4 + 3];
                *(float4*)(&xbuf[(i0 + r) * LL + j0]) = outv;
            }
        }
        __syncthreads();           // layer l+1 may read
    }

    // ---------------- Phase 3: reduction ------------------------------------
    float sum = 0.0f;
    #pragma unroll
    for (int sb = 0; sb < 2; ++sb) {
        const int blk = t + sb * 512;
        const int i0 = (blk >> 5) * 4;
        const int j0 = (blk & 31) * 4;
        #pragma unroll
        for (int r = 0; r < 4; ++r) {
            const float4 v = *(const float4*)(&xbuf[(i0 + r) * LL + j0]);
            sum += (v.x + v.y) + (v.z + v.w);
        }
    }
    #pragma unroll
    for (int off = 16; off > 0; off >>= 1)
        sum += __shfl_down(sum, off, 32);
    __syncthreads();                               // all xbuf reads done; reuse as scratch
    if (lane == 0) xbuf[wv] = sum;
    __syncthreads();
    if (t == 0) {
        float tot = 0.0f;
        for (int i = 0; i < 16; ++i) tot += xbuf[i];
        partial[bc] = tot;                         // one writer per slot: deterministic
    }
}

__global__ void finalize_kernel(const float* __restrict__ partial,
                                float* __restrict__ out)
{
    const int b = threadIdx.x;                     // 256 threads, 1 block
    float s = 0.0f;
    #pragma unroll
    for (int c = 0; c < NCH; ++c) s += partial[b * NCH + c];  // fixed order
    out[b] = expf(-s / (float)(NCH * LL * LL));
}

extern "C" void kernel_launch(void* const* d_in, const int* in_sizes, int n_in,
                              void* d_out, int out_size, void* d_ws, size_t ws_size,
                              hipStream_t stream)
{
    const float* n = (const float*)d_in[0];   // (256,128,3) f32
    const float* w = (const float*)d_in[1];   // (16,8,5)    f32
    float* out     = (float*)d_out;           // (256,)      f32
    float* partial = (float*)d_ws;            // 2048 floats scratch

    gram_stencil_kernel<<<256 * NCH, 512, 0, stream>>>(n, w, partial);
    finalize_kernel<<<1, 256, 0, stream>>>(partial, out);
}